// TempLayerMemN2N_21680994910436
// MI455X (gfx1250) — compile-verified
//
#include <hip/hip_runtime.h>

typedef __attribute__((ext_vector_type(2))) float v2f;
typedef __attribute__((ext_vector_type(4))) float v4f;
typedef __attribute__((ext_vector_type(8))) float v8f;

#define MEMN   64
#define BATCHN 32
#define LN     64
#define EN     1024
#define HN     1024
#define HOPSN  3

// ------------------------------------------------------------------
// Kernel 1: stream 512 MiB of inputs once, reducing over L.
//   X[m*BATCH+b][e] = sum_l inputs[m,b,l,e]
//   qsum[b][e]      = sum_l questions[b,l,e]
// One workgroup per (m,b) row-group; 256 threads x float4 = 1024 floats/row.
// Non-temporal loads: data is read exactly once, keep it out of caches.
// ------------------------------------------------------------------
__global__ void __launch_bounds__(256) reduce_l_kernel(
    const float* __restrict__ inputs, const float* __restrict__ questions,
    float* __restrict__ X, float* __restrict__ qsum)
{
  int blk = blockIdx.x;
  const v4f* src;
  v4f* dst;
  if (blk < MEMN * BATCHN) {
    src = (const v4f*)(inputs + (size_t)blk * LN * EN);
    dst = (v4f*)(X + (size_t)blk * EN);
  } else {
    int b = blk - MEMN * BATCHN;
    src = (const v4f*)(questions + (size_t)b * LN * EN);
    dst = (v4f*)(qsum + (size_t)b * EN);
  }
  int t = threadIdx.x;
  v4f acc = {0.f, 0.f, 0.f, 0.f};
#pragma unroll 4
  for (int l = 0; l < LN; ++l) {
    v4f v = __builtin_nontemporal_load(src + (size_t)l * (EN / 4) + t);
    acc += v;
  }
  dst[t] = acc;
}

// ------------------------------------------------------------------
// fp32 WMMA 32x32 register-blocked GEMM body (V_WMMA_F32_16X16X4_F32).
// One wave computes a 32x32 output block = 2x2 of 16x16 tiles:
//   - 4 independent accumulator chains (hides WMMA latency)
//   - each A/B fragment feeds two WMMAs (8 B of operands per WMMA)
// A-frag layout: lane<16 -> (M=lane, K=k..k+1); lane>=16 -> K=k+2..k+3.
// B-frag mirrors: half-wave selects the K row pair.
// HAS_BIAS is compile-time: straight-line epilogue, no exec-mask branches.
// ------------------------------------------------------------------
template <bool HAS_BIAS>
__device__ __forceinline__ void gemm_block32(
    const float* __restrict__ Lhs, const float* __restrict__ W,
    const float* __restrict__ bias, float* __restrict__ Out,
    int m0, int n0, int lane)
{
  const int hv  = lane >> 4;     // half-wave selector (K offset 0 or 2)
  const int l16 = lane & 15;

  const float* arow0 = Lhs + (size_t)(m0 + l16) * EN + 2 * hv;   // rows m0..m0+15
  const float* arow1 = arow0 + (size_t)16 * EN;                  // rows m0+16..m0+31
  const float* bcol0 = W + (size_t)(2 * hv) * HN + (n0 + l16);   // cols n0..n0+15
  const float* bcol1 = bcol0 + 16;                               // cols n0+16..n0+31

  const v8f vzero = {0.f, 0.f, 0.f, 0.f, 0.f, 0.f, 0.f, 0.f};
  v8f acc00 = vzero, acc01 = vzero, acc10 = vzero, acc11 = vzero;

  for (int k = 0; k < EN; k += 4) {
    v2f a0 = *(const v2f*)(arow0 + k);
    v2f a1 = *(const v2f*)(arow1 + k);
    v2f b0, b1;
    b0.x = bcol0[(size_t)k * HN];
    b0.y = bcol0[(size_t)(k + 1) * HN];
    b1.x = bcol1[(size_t)k * HN];
    b1.y = bcol1[(size_t)(k + 1) * HN];
    acc00 = __builtin_amdgcn_wmma_f32_16x16x4_f32(false, a0, false, b0, (short)0, acc00, false, false);
    acc01 = __builtin_amdgcn_wmma_f32_16x16x4_f32(false, a0, false, b1, (short)0, acc01, false, false);
    acc10 = __builtin_amdgcn_wmma_f32_16x16x4_f32(false, a1, false, b0, (short)0, acc10, false, false);
    acc11 = __builtin_amdgcn_wmma_f32_16x16x4_f32(false, a1, false, b1, (short)0, acc11, false, false);
  }

  // D layout: VGPR r -> row (tileM + r) for lanes 0-15, (tileM + 8 + r) for 16-31.
#pragma unroll
  for (int ti = 0; ti < 2; ++ti) {
#pragma unroll
    for (int tj = 0; tj < 2; ++tj) {
      v8f acc = (ti == 0) ? ((tj == 0) ? acc00 : acc01)
                          : ((tj == 0) ? acc10 : acc11);
      int col = n0 + 16 * tj + l16;
#pragma unroll
      for (int r = 0; r < 8; ++r) {
        int row = m0 + 16 * ti + r + 8 * hv;
        float v = acc[r];
        if (HAS_BIAS) v += bias[(size_t)(row >> 5) * HN + col];  // TA/TC[mem_slot][h]
        Out[(size_t)row * HN + col] = v;
      }
    }
  }
}

// Mmat = X @ A + TA,  Cmat = X @ C + TC   (2 matrices x 64 x 32 blocks = 4096 waves)
__global__ void __launch_bounds__(256) gemm_mc_kernel(
    const float* __restrict__ X,
    const float* __restrict__ Aw, const float* __restrict__ Cw,
    const float* __restrict__ TA, const float* __restrict__ TC,
    float* __restrict__ Mmat, float* __restrict__ Cmat)
{
  int wave = (int)((blockIdx.x * blockDim.x + threadIdx.x) >> 5);
  int lane = threadIdx.x & 31;
  int matId = wave >> 11;          // 0 -> A/Mmat, 1 -> C/Cmat
  int r     = wave & 2047;
  int mt = r >> 5, nt = r & 31;    // 64 m-blocks x 32 n-blocks
  const float* W    = matId ? Cw   : Aw;
  const float* bias = matId ? TC   : TA;
  float*       Out  = matId ? Cmat : Mmat;
  gemm_block32<true>(X, W, bias, Out, mt * 32, nt * 32, lane);
}

// u0 = qsum @ B  (32x1024: 1 m-block x 32 n-blocks = 32 waves), written to d_out
__global__ void __launch_bounds__(256) gemm_u0_kernel(
    const float* __restrict__ qsum, const float* __restrict__ Bw,
    float* __restrict__ u0)
{
  int wave = (int)((blockIdx.x * blockDim.x + threadIdx.x) >> 5);
  int lane = threadIdx.x & 31;
  gemm_block32<false>(qsum, Bw, nullptr, u0, 0, wave * 32, lane);
}

// ------------------------------------------------------------------
// Kernel 3: 3 memory hops, one workgroup per batch element.
//   p = softmax_j(u . m_j);  u += sum_j p_j c_j
// m/c rows are L2-resident (16 MiB total). u lives in LDS.
// ------------------------------------------------------------------
__global__ void __launch_bounds__(256) hops_kernel(
    const float* __restrict__ Mmat, const float* __restrict__ Cmat,
    float* __restrict__ u)
{
  __shared__ float u_s[HN];
  __shared__ float part[MEMN][4];
  __shared__ float p_s[MEMN];
  __shared__ float inv_s;

  int b = blockIdx.x;
  int t = threadIdx.x;
  for (int i = t; i < HN; i += 256) u_s[i] = u[(size_t)b * HN + i];
  __syncthreads();

  int j = t >> 2, sub = t & 3;   // 4 threads per memory slot j
  const float* mrow = Mmat + ((size_t)j * BATCHN + b) * HN + sub * 256;

  for (int hop = 0; hop < HOPSN; ++hop) {
    // scores[j] = u . m_j  (each thread: 256-wide partial dot)
    float s = 0.f;
#pragma unroll 8
    for (int h = 0; h < 256; ++h) s += u_s[sub * 256 + h] * mrow[h];
    part[j][sub] = s;
    __syncthreads();
    if (t < MEMN) p_s[t] = part[t][0] + part[t][1] + part[t][2] + part[t][3];
    __syncthreads();

    // softmax over 64 slots (cheap; single lane)
    if (t == 0) {
      float mx = p_s[0];
      for (int i = 1; i < MEMN; ++i) mx = fmaxf(mx, p_s[i]);
      float sum = 0.f;
      for (int i = 0; i < MEMN; ++i) { float e = __expf(p_s[i] - mx); p_s[i] = e; sum += e; }
      inv_s = 1.0f / sum;
    }
    __syncthreads();
    float inv = inv_s;

    // o[h] = sum_j p_j * c_j[h];  u += o  (each thread owns 4 h's)
    float onew[4] = {0.f, 0.f, 0.f, 0.f};
    for (int jj = 0; jj < MEMN; ++jj) {
      const float* crow = Cmat + ((size_t)jj * BATCHN + b) * HN;
      float p = p_s[jj];
#pragma unroll
      for (int q = 0; q < 4; ++q) onew[q] += p * crow[t + q * 256];
    }
#pragma unroll
    for (int q = 0; q < 4; ++q) u_s[t + q * 256] += onew[q] * inv;
    __syncthreads();
  }

  for (int i = t; i < HN; i += 256) u[(size_t)b * HN + i] = u_s[i];
}

// ------------------------------------------------------------------
extern "C" void kernel_launch(void* const* d_in, const int* in_sizes, int n_in,
                              void* d_out, int out_size, void* d_ws, size_t ws_size,
                              hipStream_t stream) {
  const float* inputs    = (const float*)d_in[0];  // [64,32,64,1024]
  const float* questions = (const float*)d_in[1];  // [32,64,1024]
  const float* Aw        = (const float*)d_in[2];  // [1024,1024]
  const float* Cw        = (const float*)d_in[3];  // [1024,1024]
  const float* Bw        = (const float*)d_in[4];  // [1024,1024]
  const float* TA        = (const float*)d_in[5];  // [64,1024]
  const float* TC        = (const float*)d_in[6];  // [64,1024]
  float* u = (float*)d_out;                        // [32,1024]

  float* ws   = (float*)d_ws;
  float* X    = ws;                                  // 2048*1024
  float* qsum = X    + (size_t)MEMN * BATCHN * EN;   // 32*1024
  float* Mmat = qsum + (size_t)BATCHN * EN;          // 2048*1024
  float* Cmat = Mmat + (size_t)MEMN * BATCHN * HN;   // 2048*1024  (~24.2 MiB total)

  // 1) L-reduction: 2048 blocks for inputs + 32 for questions
  reduce_l_kernel<<<MEMN * BATCHN + BATCHN, 256, 0, stream>>>(inputs, questions, X, qsum);

  // 2) WMMA GEMMs: 4096 waves (512 blocks of 8 waves) + 32 waves for u0
  gemm_mc_kernel<<<512, 256, 0, stream>>>(X, Aw, Cw, TA, TC, Mmat, Cmat);
  gemm_u0_kernel<<<4, 256, 0, stream>>>(qsum, Bw, u);

  // 3) hops (u0 already in d_out, updated in place)
  hops_kernel<<<BATCHN, 256, 0, stream>>>(Mmat, Cmat, u);
}